// CosineSimilarityLoss_42838003810753
// MI455X (gfx1250) — compile-verified
//
#include <hip/hip_runtime.h>
#include <math.h>

// ---------------------------------------------------------------------------
// CosineSimilarityLoss: loss = 1 - (1/B) * sum_b (1/len_b) * sum_{s<len_b} cos(o[b,s], t[b,s])
// B=32, S=2048, D=512, f32. Pure HBM-streaming reduction (~268 MB, ~11.5 us floor).
// ---------------------------------------------------------------------------

#define EPS_F 1e-8f

typedef float v2f __attribute__((ext_vector_type(2)));
typedef float v4f __attribute__((ext_vector_type(4)));
typedef float v8f __attribute__((ext_vector_type(8)));
typedef int   v4i __attribute__((vector_size(4 * sizeof(int))));  // matches builtin param

typedef __attribute__((address_space(1))) v4i* global_v4i_ptr;
typedef __attribute__((address_space(3))) v4i* lds_v4i_ptr;

constexpr int Bc = 32;
constexpr int Sc = 2048;
constexpr int Dc = 512;

constexpr int NBLOCKS       = 2048;  // multiple of 64 -> WMMA finish needs no padding
constexpr int WAVES         = 8;     // 256 threads/block
constexpr int ROWS_PER_WAVE = 4;     // 2048*8*4 = 65536 = B*S rows

#if defined(__has_builtin)
#if __has_builtin(__builtin_amdgcn_global_load_async_to_lds_b128) && \
    __has_builtin(__builtin_amdgcn_s_wait_asynccnt)
#define USE_ASYNC_LDS 1
#endif
#if __has_builtin(__builtin_amdgcn_wmma_f32_16x16x4_f32)
#define USE_WMMA_F32 1
#endif
#endif

// ---------------------------------------------------------------------------
// Kernel 1: one wave per row (b,s). Each wave streams 2*2KB per row.
// Async global->LDS double buffering: wave-private buffers, ASYNCcnt-only sync.
// ---------------------------------------------------------------------------
__global__ __launch_bounds__(256) void cos_loss_main(
    const float* __restrict__ output, const float* __restrict__ target,
    const int* __restrict__ lengths, float* __restrict__ partials)
{
  const int lane = threadIdx.x & 31;
  const int wave = threadIdx.x >> 5;
  const int waveGlobal = blockIdx.x * WAVES + wave;
  const int rowBase = waveGlobal * ROWS_PER_WAVE;

  __shared__ float wsum[WAVES];

  float acc = 0.0f;  // sum of cos/len over this wave's valid rows (uniform across lanes)

#if defined(USE_ASYNC_LDS)
  // 2 buffers x 8 waves x (512 o + 512 t) floats = 64 KB LDS
  __shared__ float stage[2][WAVES][2 * Dc];

  auto issue_copy = [&](int bi, int r) {
    float* go = const_cast<float*>(output) + (size_t)r * Dc;
    float* gt = const_cast<float*>(target) + (size_t)r * Dc;
    float* lo = &stage[bi][wave][0];
    float* lt = &stage[bi][wave][Dc];
#pragma unroll
    for (int j = 0; j < 4; ++j) {
      const int idx = (j * 32 + lane) * 4;  // float index, 16B-aligned
      __builtin_amdgcn_global_load_async_to_lds_b128(
          (global_v4i_ptr)(go + idx), (lds_v4i_ptr)(lo + idx), 0, 0);
      __builtin_amdgcn_global_load_async_to_lds_b128(
          (global_v4i_ptr)(gt + idx), (lds_v4i_ptr)(lt + idx), 0, 0);
    }
  };

  issue_copy(0, rowBase);
#pragma unroll
  for (int i = 0; i < ROWS_PER_WAVE; ++i) {
    if (i + 1 < ROWS_PER_WAVE) {
      issue_copy((i + 1) & 1, rowBase + i + 1);
      __builtin_amdgcn_s_wait_asynccnt(8);  // oldest 8 copies (current buffer) done
    } else {
      __builtin_amdgcn_s_wait_asynccnt(0);
    }

    const v4f* lo4 = (const v4f*)&stage[i & 1][wave][0];
    const v4f* lt4 = (const v4f*)&stage[i & 1][wave][Dc];
    float dot = 0.0f, no2 = 0.0f, nt2 = 0.0f;
#pragma unroll
    for (int j = 0; j < 4; ++j) {
      v4f a = lo4[j * 32 + lane];
      v4f b = lt4[j * 32 + lane];
      dot += a.x * b.x + a.y * b.y + a.z * b.z + a.w * b.w;
      no2 += a.x * a.x + a.y * a.y + a.z * a.z + a.w * a.w;
      nt2 += b.x * b.x + b.y * b.y + b.z * b.z + b.w * b.w;
    }
#else
  for (int i = 0; i < ROWS_PER_WAVE; ++i) {
    const v4f* go4 = (const v4f*)(output + (size_t)(rowBase + i) * Dc);
    const v4f* gt4 = (const v4f*)(target + (size_t)(rowBase + i) * Dc);
    float dot = 0.0f, no2 = 0.0f, nt2 = 0.0f;
#pragma unroll
    for (int j = 0; j < 4; ++j) {
      v4f a = __builtin_nontemporal_load(&go4[j * 32 + lane]);
      v4f b = __builtin_nontemporal_load(&gt4[j * 32 + lane]);
      dot += a.x * b.x + a.y * b.y + a.z * b.z + a.w * b.w;
      no2 += a.x * a.x + a.y * a.y + a.z * a.z + a.w * a.w;
      nt2 += b.x * b.x + b.y * b.y + b.z * b.z + b.w * b.w;
    }
#endif
    // wave32 butterfly: all lanes end with full row sums
#pragma unroll
    for (int off = 16; off > 0; off >>= 1) {
      dot += __shfl_xor(dot, off, 32);
      no2 += __shfl_xor(no2, off, 32);
      nt2 += __shfl_xor(nt2, off, 32);
    }

    const int r = rowBase + i;
    const int b = r >> 11;          // r / 2048
    const int s = r & (Sc - 1);     // r % 2048
    const int len = lengths[b];
    const float cosv =
        dot / (fmaxf(sqrtf(no2), EPS_F) * fmaxf(sqrtf(nt2), EPS_F));
    if (s < len) acc += cosv / (float)len;
  }

  if (lane == 0) wsum[wave] = acc;
  __syncthreads();
  if (threadIdx.x == 0) {
    float s = 0.0f;
#pragma unroll
    for (int w = 0; w < WAVES; ++w) s += wsum[w];
    partials[blockIdx.x] = s;  // deterministic: fixed order, no atomics
  }
}

// ---------------------------------------------------------------------------
// Kernel 2: single wave sums the 2048 partials on the matrix pipe.
// V_WMMA_F32_16X16X4_F32 with B = ones: D[m][n] += sum_k A[m][k] (row-sum engine).
// We sum ALL of A's entries, so the A lane<->(M,K) mapping is irrelevant; ones-B
// is layout-invariant. Total = sum_m D[m][0] = (sum of c[0..7] at lane 0) +
// (sum of c[0..7] at lane 16), per the documented C/D layout.
// ---------------------------------------------------------------------------
__global__ __launch_bounds__(32) void cos_loss_finish(
    const float* __restrict__ partials, float* __restrict__ out)
{
  const int lane = threadIdx.x;
  float total;

#if defined(USE_WMMA_F32)
  v8f c = {};                 // f32 accumulator, 16x16
  v2f ones; ones.x = 1.0f; ones.y = 1.0f;
  const v2f* p2 = (const v2f*)partials;
  for (int it = 0; it < NBLOCKS / 64; ++it) {   // 64 partials per WMMA
    v2f a = p2[it * 32 + lane];
    c = __builtin_amdgcn_wmma_f32_16x16x4_f32(
        /*neg_a=*/false, a, /*neg_b=*/false, ones,
        /*c_mod=*/(short)0, c, /*reuse_a=*/false, /*reuse_b=*/false);
  }
  float s = c[0] + c[1] + c[2] + c[3] + c[4] + c[5] + c[6] + c[7];
  total = __shfl(s, 0, 32) + __shfl(s, 16, 32);   // M=0..7 (lane 0) + M=8..15 (lane 16), N=0
#else
  float s = 0.0f;
  for (int i = lane; i < NBLOCKS; i += 32) s += partials[i];
#pragma unroll
  for (int off = 16; off > 0; off >>= 1) s += __shfl_xor(s, off, 32);
  total = s;
#endif

  if (lane == 0) out[0] = 1.0f - total / (float)Bc;
}

// ---------------------------------------------------------------------------
extern "C" void kernel_launch(void* const* d_in, const int* in_sizes, int n_in,
                              void* d_out, int out_size, void* d_ws, size_t ws_size,
                              hipStream_t stream) {
  const float* output  = (const float*)d_in[0];
  const float* target  = (const float*)d_in[1];
  const int*   lengths = (const int*)d_in[2];
  float* partials = (float*)d_ws;              // NBLOCKS * 4 = 8 KB scratch
  float* out      = (float*)d_out;

  cos_loss_main<<<NBLOCKS, 256, 0, stream>>>(output, target, lengths, partials);
  cos_loss_finish<<<1, 32, 0, stream>>>(partials, out);
}